// VQGAN_45028437131516
// MI455X (gfx1250) — compile-verified
//
#include <hip/hip_runtime.h>
#include <hip/hip_bf16.h>
#include <math.h>

// ---------------------------------------------------------------------------
// VQ-VAE vector quantization for MI455X (gfx1250, wave32, WMMA + TDM).
//   distances GEMM (8192x16384x256, fp32) fused with argmin via
//   V_WMMA_F32_16X16X4_F32; codes = M rows (A tiles DMA'd into LDS by the
//   Tensor Data Mover with hardware row padding, double buffered), tokens =
//   N cols (B operand register-resident, contiguous in z for fixed channel).
// ---------------------------------------------------------------------------

typedef __attribute__((ext_vector_type(2))) float    v2f;
typedef __attribute__((ext_vector_type(8))) float    v8f;
typedef __attribute__((ext_vector_type(4))) unsigned u32x4;
typedef __attribute__((ext_vector_type(8))) int      i32x8;
typedef __attribute__((ext_vector_type(4))) int      i32x4;

#define N_CODES   16384
#define EMBED_DIM 256
#define N_TOK     8192            // 4*8*16*16 tokens
#define Z_ELEMS   2097152         // 4*256*8*16*16
#define KSPLIT    4
#define KPER      (N_CODES / KSPLIT)
#define CHUNK     32              // codes staged in LDS per iteration
#define LDS_PITCH 260             // 256 + 4 pad -> conflict-free b64 frags
#define BUFSZ     (CHUNK * LDS_PITCH)

// workspace layout (bytes)
#define WS_ENORM  0               // float[16384]
#define WS_HIST   65536           // uint [16384]
#define WS_IDX    131072          // int  [8192]
#define WS_CVAL   163840          // float[KSPLIT*8192]
#define WS_CIDX   294912          // int  [KSPLIT*8192]
#define WS_PART   425984          // float[8192] commitment partials
// total 458752 bytes

// output layout (floats, concatenated tuple)
#define OUT_ST    0
#define OUT_ENC   2097152
#define OUT_LOSS  2105344
#define OUT_PERP  2105345

#if defined(__has_builtin)
#if __has_builtin(__builtin_amdgcn_tensor_load_to_lds) && \
    __has_builtin(__builtin_amdgcn_s_wait_tensorcnt)
#define USE_TDM 1
#endif
#endif

// ---------------------------------------------------------------------------
__global__ void vq_zero(unsigned* __restrict__ hist) {
  int i = blockIdx.x * 256 + threadIdx.x;
  if (i < N_CODES) hist[i] = 0u;
}

// ||e_k||^2 per code: one wave per code, coalesced, shuffle reduce.
__global__ void vq_enorm(const float* __restrict__ emb, float* __restrict__ enorm) {
  int wave = threadIdx.x >> 5, lane = threadIdx.x & 31;
  int code = blockIdx.x * 8 + wave;
  const float* row = emb + (size_t)code * EMBED_DIM;
  float s = 0.f;
#pragma unroll
  for (int k = 0; k < 8; ++k) { float v = row[lane + 32 * k]; s += v * v; }
#pragma unroll
  for (int off = 16; off; off >>= 1) s += __shfl_xor(s, off, 32);
  if (lane == 0) enorm[code] = s;
}

#ifdef USE_TDM
// DMA one CHUNKx256 f32 tile of E into LDS via the Tensor Data Mover.
// D# group0: count=1 | lds_addr | global byte addr [120:64] | type=2.
// D# group1: data_size=4B, pad_enable, pad_interval=7 (every 256 DWORDs),
//            pad_amount=3 (4 DWORDs) -> LDS row pitch 260 floats;
//            tensor_dim0=256, tensor_dim1=N_CODES, tile=256xCHUNK, stride=256.
// Groups 2/3 unused (2D tensor) -> zero-filled; clang-23 arity is 6 args.
__device__ __forceinline__ void tdm_load_chunk(const float* gsrc, unsigned lds_byte) {
  unsigned long long ga = (unsigned long long)(uintptr_t)gsrc;
  u32x4 g0;
  g0[0] = 1u;                                            // count=1, user desc
  g0[1] = lds_byte;                                      // lds_addr (bytes)
  g0[2] = (unsigned)(ga & 0xFFFFFFFFu);                  // global_addr[31:0]
  g0[3] = (unsigned)((ga >> 32) & 0x1FFFFFFu) | (2u << 30);  // [56:32] | type=2
  i32x8 g1;
  g1[0] = (2 << 16) | (1 << 20) | (7 << 22) | (3 << 25); // 4B, pad 4dw/256dw
  g1[1] = (int)(256u << 16);                             // tensor_dim0 lo16
  g1[2] = (int)((unsigned)N_CODES << 16);                // dim0 hi | dim1 lo
  g1[3] = (int)(((unsigned)N_CODES >> 16) | (256u << 16)); // dim1 hi | tile_dim0
  g1[4] = (int)CHUNK;                                    // tile_dim1 | tile_dim2=0
  g1[5] = 256;                                           // tensor_dim0_stride lo
  g1[6] = 0;
  g1[7] = 0;
  i32x4 zz4 = {0, 0, 0, 0};
  i32x8 zz8 = {0, 0, 0, 0, 0, 0, 0, 0};
  __builtin_amdgcn_tensor_load_to_lds(g0, g1, zz4, zz4, zz8, 0);
}
#endif

// ---------------------------------------------------------------------------
// Main fused GEMM+argmin. Grid: dim3(64, KSPLIT), block 256 (8 waves).
// Block x owns tokens [x*128, x*128+128); wave w owns 16 of them.
// Block y streams codes [y*KPER, (y+1)*KPER) through LDS in CHUNK rows.
__global__ void __launch_bounds__(256)
vq_argmin(const float* __restrict__ z, const float* __restrict__ emb,
          const float* __restrict__ enorm,
          float* __restrict__ cval, int* __restrict__ cidx) {
  __shared__ __align__(16) float ldsA[2][BUFSZ];

  const int tid   = threadIdx.x;
  const int wave  = tid >> 5, lane = tid & 31;
  const int lrow  = lane & 15;           // token-in-wave / A-row-in-tile
  const int khalf = (lane >> 4) << 1;    // K offset 0 (lanes 0-15) or 2 (16-31)
  const int split = blockIdx.y;
  const int n0    = blockIdx.x * 128 + wave * 16;

  // --- Stage B operand (this wave's 16 tokens x 256 channels) in VGPRs. ---
  // token n -> z flat index b*524288 + c*2048 + t*256 + hw  (C stride 2048)
  const int n  = n0 + lrow;
  const int bt = n >> 8, hw = n & 255;
  const int bb = bt >> 3, tt = bt & 7;
  const size_t zbase = (size_t)bb * 524288 + (size_t)tt * 256 + (size_t)hw;
  v2f bfrag[64];                         // 128 VGPRs/lane, K-resident
#pragma unroll
  for (int s = 0; s < 64; ++s) {
    const int c = 4 * s + khalf;
    bfrag[s].x = z[zbase + (size_t)c * 2048];
    bfrag[s].y = z[zbase + (size_t)(c + 1) * 2048];
  }

  float minv = 3.4e38f;
  int   mini = 0;
  const int kbeg = split * KPER, kend = kbeg + KPER;
  int buf = 0;

#ifdef USE_TDM
  const unsigned lds_off0 = (unsigned)(uintptr_t)&ldsA[0][0]; // low 32 bits of
  const unsigned lds_off1 = (unsigned)(uintptr_t)&ldsA[1][0]; // flat = LDS offs
  if (wave == 0) tdm_load_chunk(emb + (size_t)kbeg * EMBED_DIM, lds_off0);
#endif

  for (int m0 = kbeg; m0 < kend; m0 += CHUNK) {
#ifdef USE_TDM
    if (wave == 0) __builtin_amdgcn_s_wait_tensorcnt((short)0);
    __syncthreads();                     // chunk m0 visible; other buf free
    if (wave == 0 && m0 + CHUNK < kend)
      tdm_load_chunk(emb + (size_t)(m0 + CHUNK) * EMBED_DIM,
                     buf ? lds_off0 : lds_off1);
#else
    float4 tmp[8];
#pragma unroll
    for (int k = 0; k < 8; ++k) {
      const int idx = tid + (k << 8);
      const int r = idx >> 6, c4 = idx & 63;
      tmp[k] = *(const float4*)(emb + (size_t)(m0 + r) * EMBED_DIM + c4 * 4);
    }
    __syncthreads();
#pragma unroll
    for (int k = 0; k < 8; ++k) {
      const int idx = tid + (k << 8);
      const int r = idx >> 6, c4 = idx & 63;
      *(float4*)&ldsA[0][r * LDS_PITCH + c4 * 4] = tmp[k];
    }
    __syncthreads();
#endif

#pragma unroll
    for (int half = 0; half < 2; ++half) {
      const float* Arow =
          &ldsA[buf][(half * 16 + lrow) * LDS_PITCH + khalf];
      // ||e||^2 for this lane's 8 rows: two broadcast float4 from L2
      const int lds_r = half * 16 + ((lane >> 4) << 3);
      const int mglob = m0 + lds_r;
      const float4 en0 = *(const float4*)(enorm + mglob);
      const float4 en1 = *(const float4*)(enorm + mglob + 4);
      v8f acc = {};
#pragma unroll
      for (int s = 0; s < 64; ++s) {
        const v2f a = *(const v2f*)(Arow + 4 * s);  // ds_load_b64, bank-clean
        acc = __builtin_amdgcn_wmma_f32_16x16x4_f32(
            false, a, false, bfrag[s], (short)0, acc, false, false);
      }
      // D layout: VGPR v -> code row (half*16 + v) lanes 0-15, (+8) lanes 16-31
      const float en[8] = {en0.x, en0.y, en0.z, en0.w, en1.x, en1.y, en1.z, en1.w};
#pragma unroll
      for (int v = 0; v < 8; ++v) {
        const float d = en[v] - 2.0f * acc[v];
        if (d < minv) { minv = d; mini = mglob + v; }
      }
    }
#ifdef USE_TDM
    buf ^= 1;
#endif
  }

  // fold lane n+16 (codes +8 per tile) into lane n, prefer smaller index on tie
  const float ov = __shfl_xor(minv, 16, 32);
  const int   oi = __shfl_xor(mini, 16, 32);
  if (ov < minv || (ov == minv && oi < mini)) { minv = ov; mini = oi; }
  if (lane < 16) {
    const int tok = n0 + lane;
    cval[split * N_TOK + tok] = minv;
    cidx[split * N_TOK + tok] = mini;
  }
}

// ---------------------------------------------------------------------------
__global__ void vq_combine(const float* __restrict__ cval, const int* __restrict__ cidx,
                           int* __restrict__ idxout, float* __restrict__ enc_out,
                           unsigned* __restrict__ hist) {
  const int nn = blockIdx.x * 256 + threadIdx.x;
  if (nn >= N_TOK) return;
  float mv = cval[nn];
  int   mi = cidx[nn];
#pragma unroll
  for (int s = 1; s < KSPLIT; ++s) {
    const float v = cval[s * N_TOK + nn];
    const int   i = cidx[s * N_TOK + nn];
    if (v < mv || (v == mv && i < mi)) { mv = v; mi = i; }
  }
  idxout[nn] = mi;
  enc_out[nn] = (float)mi;
  atomicAdd(&hist[mi], 1u);              // integer atomics -> deterministic
}

// gather codebook rows, straight-through output, per-block commitment partial
__global__ void vq_gather(const float* __restrict__ z, const float* __restrict__ emb,
                          const int* __restrict__ idx, float* __restrict__ out_st,
                          float* __restrict__ partials) {
  __shared__ float red[8];
  const int i   = blockIdx.x * 256 + threadIdx.x;   // linear z index
  const int hw  = threadIdx.x;
  const int blk = blockIdx.x;
  const int t = blk & 7, c = (blk >> 3) & 255, b = blk >> 11;
  const int nn = (((b << 3) | t) << 8) | hw;
  const int code = idx[nn];
  const float e  = emb[(size_t)code * EMBED_DIM + c];
  const float zv = z[i];
  out_st[i] = (e - zv) + zv;             // match reference (emb - z) + z rounding
  const float d = zv - e;
  float s = d * d;
#pragma unroll
  for (int off = 16; off; off >>= 1) s += __shfl_xor(s, off, 32);
  const int lane = threadIdx.x & 31, wv = threadIdx.x >> 5;
  if (lane == 0) red[wv] = s;
  __syncthreads();
  if (threadIdx.x == 0) {
    float tot = 0.f;
#pragma unroll
    for (int w = 0; w < 8; ++w) tot += red[w];
    partials[blk] = tot;                 // deterministic per-block partial
  }
}

// perplexity + commitment loss (single block, fixed-order reductions)
__global__ void vq_final(const unsigned* __restrict__ hist,
                         const float* __restrict__ partials,
                         float* __restrict__ out) {
  __shared__ float red[256];
  const int tid = threadIdx.x;
  float s = 0.f;
  for (int k = tid; k < N_CODES; k += 256) {
    const float p = (float)hist[k] * (1.0f / (float)N_TOK);
    s += p * logf(p + 1e-10f);
  }
  float ls = 0.f;
  for (int k = tid; k < N_TOK; k += 256) ls += partials[k];

  red[tid] = s;
  __syncthreads();
  for (int off = 128; off; off >>= 1) {
    if (tid < off) red[tid] += red[tid + off];
    __syncthreads();
  }
  const float perp_sum = red[0];
  __syncthreads();
  red[tid] = ls;
  __syncthreads();
  for (int off = 128; off; off >>= 1) {
    if (tid < off) red[tid] += red[tid + off];
    __syncthreads();
  }
  if (tid == 0) {
    out[OUT_PERP] = expf(-perp_sum);
    out[OUT_LOSS] = 0.25f * red[0] / (float)Z_ELEMS;
  }
}

// ---------------------------------------------------------------------------
extern "C" void kernel_launch(void* const* d_in, const int* in_sizes, int n_in,
                              void* d_out, int out_size, void* d_ws, size_t ws_size,
                              hipStream_t stream) {
  const float* z   = (const float*)d_in[0];   // [4,256,8,16,16]
  const float* emb = (const float*)d_in[1];   // [16384,256]
  float* out = (float*)d_out;
  char*  ws  = (char*)d_ws;

  float*    enorm = (float*)(ws + WS_ENORM);
  unsigned* hist  = (unsigned*)(ws + WS_HIST);
  int*      idx   = (int*)(ws + WS_IDX);
  float*    cval  = (float*)(ws + WS_CVAL);
  int*      cidx  = (int*)(ws + WS_CIDX);
  float*    part  = (float*)(ws + WS_PART);

  vq_zero  <<<N_CODES / 256, 256, 0, stream>>>(hist);
  vq_enorm <<<N_CODES / 8,   256, 0, stream>>>(emb, enorm);
  vq_argmin<<<dim3(64, KSPLIT), 256, 0, stream>>>(z, emb, enorm, cval, cidx);
  vq_combine<<<N_TOK / 256,  256, 0, stream>>>(cval, cidx, idx, out + OUT_ENC, hist);
  vq_gather<<<Z_ELEMS / 256, 256, 0, stream>>>(z, emb, idx, out + OUT_ST, part);
  vq_final <<<1, 256, 0, stream>>>(hist, part, out);
}